// qg_flux_9010841387192
// MI455X (gfx1250) — compile-verified
//
#include <hip/hip_runtime.h>
#include <stdint.h>
#include <stddef.h>

#ifndef __has_builtin
#define __has_builtin(x) 0
#endif

// ---------------------------------------------------------------------------
// gfx1250 async global->LDS copy path (CDNA5 Tensor/Async data movement).
// Probe round 1 showed the builtin takes (global int*, local int*, imm, imm).
// Falls back to plain LDS stores if the builtin is unavailable.
// ---------------------------------------------------------------------------
#if defined(__HIP_DEVICE_COMPILE__) && __has_builtin(__builtin_amdgcn_global_load_async_to_lds_b32)
#define QG_ASYNC_LDS 1
#else
#define QG_ASYNC_LDS 0
#endif

__device__ __forceinline__ void lds_copy_f32(const float* gsrc, float* ldst) {
#if QG_ASYNC_LDS
  __builtin_amdgcn_global_load_async_to_lds_b32(
      (__attribute__((address_space(1))) int*)(uintptr_t)gsrc,
      (__attribute__((address_space(3))) int*)ldst,
      0, 0);
#else
  *ldst = *gsrc;
#endif
}

__device__ __forceinline__ void lds_copy_wait() {
#if QG_ASYNC_LDS
#if __has_builtin(__builtin_amdgcn_s_wait_asynccnt)
  __builtin_amdgcn_s_wait_asynccnt(0);
#else
  asm volatile("s_wait_asynccnt 0" ::: "memory");
#endif
#endif
}

#define QG_B 16  // batch

__device__ __forceinline__ int clampi(int v, int lo, int hi) {
  return v < lo ? lo : (v > hi ? hi : v);
}

// ---------------------------------------------------------------------------
// Fused double-Jacobi relax: performs TWO reference relax() sweeps per launch.
//   new[i,j] = c*(up+dn+lf+rt - q[i,j]) interior, 0 on boundary.
// Block: 256 threads, 32x32 output tile, radius-2 halo staged via async LDS.
// psi_is_zero=1 emulates psis[k]=zeros before the first sweep.
// ---------------------------------------------------------------------------
__global__ __launch_bounds__(256)
void relax2_kernel(const float* __restrict__ psi, const float* __restrict__ q,
                   float* __restrict__ dst, int S, float c, int psi_is_zero) {
  const int S2 = S * S;
  const int b = blockIdx.z;
  const int base_r = blockIdx.y * 32;
  const int base_c = blockIdx.x * 32;
  const float* psiB = psi + (size_t)b * S2;
  const float* qB = q + (size_t)b * S2;
  float* dB = dst + (size_t)b * S2;

  __shared__ float sp[36 * 36];  // psi halo: rows base_r-2..base_r+33
  __shared__ float sq[34 * 34];  // q halo:   rows base_r-1..base_r+32
  __shared__ float st[34 * 34];  // iteration-1 result, same window as sq

  const int t = threadIdx.x;

  if (!psi_is_zero) {
    for (int i = t; i < 36 * 36; i += 256) {
      int lr = i / 36, lc = i % 36;
      int gr = clampi(base_r - 2 + lr, 0, S - 1);
      int gc = clampi(base_c - 2 + lc, 0, S - 1);
      lds_copy_f32(psiB + gr * S + gc, &sp[i]);
    }
  }
  for (int i = t; i < 34 * 34; i += 256) {
    int lr = i / 34, lc = i % 34;
    int gr = clampi(base_r - 1 + lr, 0, S - 1);
    int gc = clampi(base_c - 1 + lc, 0, S - 1);
    lds_copy_f32(qB + gr * S + gc, &sq[i]);
  }
  lds_copy_wait();
  __syncthreads();

  // Iteration 1 over the 34x34 window (global rows base_r-1..base_r+32)
  for (int i = t; i < 34 * 34; i += 256) {
    int lr = i / 34, lc = i % 34;
    int gr = base_r - 1 + lr, gc = base_c - 1 + lc;
    float v = 0.0f;
    if (gr > 0 && gr < S - 1 && gc > 0 && gc < S - 1) {
      float up, dn, lf, rt;
      if (psi_is_zero) {
        up = dn = lf = rt = 0.0f;
      } else {
        // sp index of global (gr',gc') = (gr'-base_r+2, gc'-base_c+2)
        up = sp[(lr + 0) * 36 + (lc + 1)];
        dn = sp[(lr + 2) * 36 + (lc + 1)];
        lf = sp[(lr + 1) * 36 + (lc + 0)];
        rt = sp[(lr + 1) * 36 + (lc + 2)];
      }
      v = c * ((((up + dn) + lf) + rt) - sq[i]);
    }
    st[i] = v;
  }
  __syncthreads();

  // Iteration 2: 32x32 outputs (global rows base_r..base_r+31)
  for (int i = t; i < 32 * 32; i += 256) {
    int lr = i / 32, lc = i % 32;
    int gr = base_r + lr, gc = base_c + lc;
    if (gr >= S || gc >= S) continue;
    float v = 0.0f;
    if (gr > 0 && gr < S - 1 && gc > 0 && gc < S - 1) {
      float up = st[(lr + 0) * 34 + (lc + 1)];
      float dn = st[(lr + 2) * 34 + (lc + 1)];
      float lf = st[(lr + 1) * 34 + (lc + 0)];
      float rt = st[(lr + 1) * 34 + (lc + 2)];
      v = c * ((((up + dn) + lf) + rt) - sq[(lr + 1) * 34 + (lc + 1)]);
    }
    dB[gr * S + gc] = v;
  }
}

// ---------------------------------------------------------------------------
// init: psi0 = PSIGUESS, q0 = Q * h0^2
// ---------------------------------------------------------------------------
__global__ __launch_bounds__(256)
void init_kernel(const float* __restrict__ psig, const float* __restrict__ q,
                 float* __restrict__ psi0, float* __restrict__ q0, float h2, int tot) {
  int idx = blockIdx.x * 256 + threadIdx.x;
  if (idx >= tot) return;
  psi0[idx] = psig[idx];
  q0[idx] = q[idx] * h2;
}

// ---------------------------------------------------------------------------
// rescale: q_coarse[I,J] = 4*(d*psi[2I,2J]-psi[2I-1,2J]-psi[2I+1,2J]
//                             -psi[2I,2J-1]-psi[2I,2J+1]+q[2I,2J]) interior, else 0
// ---------------------------------------------------------------------------
__global__ __launch_bounds__(256)
void rescale_kernel(const float* __restrict__ psi, const float* __restrict__ q,
                    float* __restrict__ qc, int Sf, int Sc, float d) {
  int idx = blockIdx.x * 256 + threadIdx.x;
  int tot = QG_B * Sc * Sc;
  if (idx >= tot) return;
  int b = idx / (Sc * Sc);
  int r = (idx / Sc) % Sc;
  int c = idx % Sc;
  int no = (Sf - 5) / 2 + 1;
  float v = 0.0f;
  if (r >= 1 && r <= no && c >= 1 && c <= no) {
    const float* pB = psi + (size_t)b * Sf * Sf;
    const float* qB = q + (size_t)b * Sf * Sf;
    int fr = 2 * r, fc = 2 * c;
    float ctr = pB[fr * Sf + fc];
    float up = pB[(fr - 1) * Sf + fc];
    float dn = pB[(fr + 1) * Sf + fc];
    float lf = pB[fr * Sf + fc - 1];
    float rt = pB[fr * Sf + fc + 1];
    float qv = qB[fr * Sf + fc];
    v = 4.0f * (((((d * ctr - up) - dn) - lf) - rt) + qv);
  }
  qc[idx] = v;
}

// ---------------------------------------------------------------------------
// addup: fine += bilinear_up(coarse);  Sf = 2*Sc - 1 exactly.
// ---------------------------------------------------------------------------
__global__ __launch_bounds__(256)
void addup_kernel(float* __restrict__ fine, const float* __restrict__ coarse,
                  int Sf, int Sc) {
  int idx = blockIdx.x * 256 + threadIdx.x;
  int tot = QG_B * Sf * Sf;
  if (idx >= tot) return;
  int b = idx / (Sf * Sf);
  int r = (idx / Sf) % Sf;
  int c = idx % Sf;
  const float* cB = coarse + (size_t)b * Sc * Sc;
  int r2 = r >> 1, c2 = c >> 1;
  int ro = r & 1, co = c & 1;
  if (r2 > Sc - 2) r2 = Sc - 1;  // even last row maps directly
  if (c2 > Sc - 2) c2 = Sc - 1;
  float a = cB[r2 * Sc + c2];
  float rowv = ro ? (a * 0.5f + cB[(r2 + 1) * Sc + c2] * 0.5f) : a;
  float v;
  if (co) {
    float a2 = cB[r2 * Sc + c2 + 1];
    float rowv2 = ro ? (a2 * 0.5f + cB[(r2 + 1) * Sc + c2 + 1] * 0.5f) : a2;
    v = rowv * 0.5f + rowv2 * 0.5f;
  } else {
    v = rowv;
  }
  fine[idx] = fine[idx] + v;
}

// ---------------------------------------------------------------------------
// 5-point Laplacian, dx=dy=2^-9 so 1/dx^2 = 262144 exactly.
// ---------------------------------------------------------------------------
__global__ __launch_bounds__(256)
void lap_kernel(const float* __restrict__ A, float* __restrict__ out, int S) {
  int idx = blockIdx.x * 256 + threadIdx.x;
  int tot = QG_B * S * S;
  if (idx >= tot) return;
  int b = idx / (S * S);
  int r = (idx / S) % S;
  int c = idx % S;
  float v = 0.0f;
  if (r > 0 && r < S - 1 && c > 0 && c < S - 1) {
    const float* aB = A + (size_t)b * S * S;
    int i0 = r * S + c;
    v = (aB[i0 - 1] + aB[i0 + 1]) * 262144.0f +
        (aB[i0 - S] + aB[i0 + S]) * 262144.0f - aB[i0] * 1048576.0f;
  }
  out[idx] = v;
}

// ---------------------------------------------------------------------------
// Fused final: J = arakawa(PSI,Q), Z4 = lap(Z2) inline, combine to QFLUX.
// ---------------------------------------------------------------------------
__global__ __launch_bounds__(256)
void final_kernel(const float* __restrict__ PSI, const float* __restrict__ Qin,
                  const float* __restrict__ Z, const float* __restrict__ Z2,
                  const float* __restrict__ CURL, float* __restrict__ out, int S) {
  int idx = blockIdx.x * 256 + threadIdx.x;
  int tot = QG_B * S * S;
  if (idx >= tot) return;
  int b = idx / (S * S);
  int r = (idx / S) % S;
  int c = idx % S;
  float v = 0.0f;
  if (r > 0 && r < S - 1 && c > 0 && c < S - 1) {
    size_t bo = (size_t)b * S * S;
    const float* A = PSI + bo;
    const float* Bq = Qin + bo;
    const float* z2 = Z2 + bo;
    int i0 = r * S + c;
    float Amm = A[i0 - S - 1], Am0 = A[i0 - S], Amp = A[i0 - S + 1];
    float A0m = A[i0 - 1], A0p = A[i0 + 1];
    float Apm = A[i0 + S - 1], Ap0 = A[i0 + S], App = A[i0 + S + 1];
    float Bmm = Bq[i0 - S - 1], Bm0 = Bq[i0 - S], Bmp = Bq[i0 - S + 1];
    float B0m = Bq[i0 - 1], B0p = Bq[i0 + 1];
    float Bpm = Bq[i0 + S - 1], Bp0 = Bq[i0 + S], Bpp = Bq[i0 + S + 1];
    float t1 = (A0m - Am0) * Bmm;
    float t2 = (((Amm + A0m) - Amp) - A0p) * Bm0;
    float t3 = (Am0 - A0p) * Bmp;
    float t4 = (((Apm + Ap0) - Amm) - Am0) * B0m;
    float t5 = (((Am0 + Amp) - Ap0) - App) * B0p;
    float t6 = (Ap0 - A0m) * Bpm;
    float t7 = (((A0p + App) - A0m) - Apm) * Bp0;
    float t8 = (A0p - Ap0) * Bpp;
    float J = ((((((t1 + t2) + t3) + t4) + t5) + t6) + t7) + t8;
    float j = J / 4.57763671875e-05f;  // 12*DX*DY, exact in fp32
    float zc = Z[bo + i0];
    float z2c = z2[i0];
    float z4 = (z2[i0 - 1] + z2[i0 + 1]) * 262144.0f +
               (z2[i0 - S] + z2[i0 + S]) * 262144.0f - z2c * 1048576.0f;
    float curl = CURL[i0];  // (1,S,S) broadcast over batch
    float base = ((((-j) - 1e-6f * zc) + 1e-8f * z2c) - 2e-11f * z4) + curl;
    v = base - 256.0f * (A0p - A0m);  // 0.5/DX = 256 exactly
  }
  out[idx] = v;
}

// ---------------------------------------------------------------------------
// Host driver: full multigrid V-cycle sequence exactly mirroring reference,
// replayed deterministically each call. 64 launches, graph-capture safe.
// ---------------------------------------------------------------------------
extern "C" void kernel_launch(void* const* d_in, const int* in_sizes, int n_in,
                              void* d_out, int out_size, void* d_ws, size_t ws_size,
                              hipStream_t stream) {
  (void)in_sizes; (void)n_in; (void)out_size; (void)ws_size;
  const float* PSIG = (const float*)d_in[1];
  const float* QIN = (const float*)d_in[2];
  const float* CURL = (const float*)d_in[3];

  int S[8];
  S[0] = 513;
  for (int k = 1; k < 8; ++k) S[k] = (S[k - 1] + 1) / 2;  // 257,129,65,33,17,9,5

  float* ws = (float*)d_ws;
  float* psiBuf[8][2];
  float* qBuf[8];
  size_t off = 0;
  for (int k = 0; k < 8; ++k) {
    size_t n = (size_t)QG_B * S[k] * S[k];
    psiBuf[k][0] = ws + off; off += n;
    psiBuf[k][1] = ws + off; off += n;
    qBuf[k] = ws + off; off += n;
  }

  float cF[8], dF[8];
  for (int k = 0; k < 8; ++k) {
    double h = 0.25 / (double)(1 << (7 - k));  // HS[k]
    double fh2 = 1600.0 * h * h;
    cF[k] = (float)(1.0 / (4.0 + fh2));
    dF[k] = (float)(4.0 + fh2);
  }

  float* outPSI = (float*)d_out;
  float* outQF = outPSI + (size_t)QG_B * 513 * 513;

  const size_t n0 = (size_t)QG_B * 513 * 513;
  const int blk0 = (int)((n0 + 255) / 256);

  init_kernel<<<blk0, 256, 0, stream>>>(PSIG, QIN, psiBuf[0][0], qBuf[0],
                                        3.814697265625e-06f, (int)n0);

  int cur[8] = {0, 0, 0, 0, 0, 0, 0, 0};

  auto relax2 = [&](int k, const float* src, float* dst, int zero) {
    int s = S[k];
    dim3 g((s + 31) / 32, (s + 31) / 32, QG_B);
    relax2_kernel<<<g, 256, 0, stream>>>(src, qBuf[k], dst, s, cF[k], zero);
  };

  for (int cyc = 0; cyc < 2; ++cyc) {
    // ---- downward pass ----
    for (int k = 0; k < 8; ++k) {
      if (k == 0) {
        relax2(0, psiBuf[0][cur[0]], psiBuf[0][cur[0] ^ 1], 0);
        cur[0] ^= 1;
      } else {
        relax2(k, psiBuf[k][0], psiBuf[k][0], 1);  // psis[k]=0 then 2 sweeps
        cur[k] = 0;
      }
      if (k < 7) {
        int sc = S[k + 1];
        size_t tot = (size_t)QG_B * sc * sc;
        int blocks = (int)((tot + 255) / 256);
        rescale_kernel<<<blocks, 256, 0, stream>>>(psiBuf[k][cur[k]], qBuf[k],
                                                   qBuf[k + 1], S[k], sc, dF[k]);
      }
    }
    // ---- upward pass (note reference uses loop index kk for prolongation) ----
    for (int kk = 0; kk < 8; ++kk) {
      int lvl = 7 - kk;
      bool last = (cyc == 1 && kk == 7);
      float* dst = last ? outPSI : psiBuf[lvl][cur[lvl] ^ 1];
      relax2(lvl, psiBuf[lvl][cur[lvl]], dst, 0);
      if (!last) cur[lvl] ^= 1;
      if (kk > 0) {
        int sf = S[kk - 1], sc = S[kk];
        size_t tot = (size_t)QG_B * sf * sf;
        int blocks = (int)((tot + 255) / 256);
        addup_kernel<<<blocks, 256, 0, stream>>>(psiBuf[kk - 1][cur[kk - 1]],
                                                 psiBuf[kk][cur[kk]], sf, sc);
      }
    }
  }

  // Z = lap(PSI), Z2 = lap(Z); level-0 scratch buffers are free now.
  float* Z = psiBuf[0][0];
  float* Z2 = psiBuf[0][1];
  lap_kernel<<<blk0, 256, 0, stream>>>(outPSI, Z, 513);
  lap_kernel<<<blk0, 256, 0, stream>>>(Z, Z2, 513);
  final_kernel<<<blk0, 256, 0, stream>>>(outPSI, QIN, Z, Z2, CURL, outQF, 513);
}